// VectorQuantizerEMA_43233140802032
// MI455X (gfx1250) — compile-verified
//
#include <hip/hip_runtime.h>
#include <hip/hip_bf16.h>
#include <stdint.h>

#define K_CLUSTERS 1024
#define DIMS 256
#define N_VECS 32768
#define DECAY 0.99f
#define ONE_MINUS_DECAY 0.01f
#define EPSV 1e-5f
#define FP_SCALE 1048576.0f
#define INV_FP_SCALE (1.0f/1048576.0f)

typedef __attribute__((ext_vector_type(2))) float v2f;
typedef __attribute__((ext_vector_type(4))) float v4f;
typedef __attribute__((ext_vector_type(8))) float v8f;

// ---- workspace layout (bytes) ----
#define WS_NH     0                                   // float[1024]  : -0.5*||e_k||^2
#define WS_BPERM  4096                                // float2[64][64][32] : WMMA-layout B (1 MB)
#define WS_IDX    (WS_BPERM + 64*64*32*2*4)           // int[32768]
#define WS_COUNTS (WS_IDX + 32768*4)                  // uint[1024]
#define WS_ACC    (WS_COUNTS + 4096)                  // u64[262144] fixed-point EMA accum (2 MB)

// ---- output offsets (floats, concatenated in reference return order) ----
#define OUT_ENC   0ull          // [32768,1024]
#define OUT_QUANT 33554432ull   // [32,256,32,32]
#define OUT_EMB   41943040ull   // [1024,256]
#define OUT_CS    42205184ull   // [1024]
#define OUT_EMA   42206208ull   // [1024,256]

// -0.5*||e_k||^2 per code; one wave per row.
__global__ void nh_e2_kernel(const float* __restrict__ E, float* __restrict__ nh) {
    int lane = threadIdx.x & 31;
    int row  = blockIdx.x * 8 + (threadIdx.x >> 5);
    const float* e = E + row * DIMS;
    float s = 0.f;
#pragma unroll
    for (int i = 0; i < DIMS / 32; ++i) { float v = e[lane + i * 32]; s += v * v; }
#pragma unroll
    for (int off = 16; off >= 1; off >>= 1) s += __shfl_xor(s, off, 32);
    if (lane == 0) nh[row] = -0.5f * s;
}

// Permute embeddings into the exact per-lane B-operand layout of V_WMMA_F32_16X16X4_F32.
// B chunk is (4 x 16) = E^T slice: v0 = B[k0][n] (lanes 0-15), B[k0+2][n] (lanes 16-31); v1 = +1.
__global__ void bperm_kernel(const float* __restrict__ E, v2f* __restrict__ bp) {
    int t     = blockIdx.x * blockDim.x + threadIdx.x;   // 131072 threads
    int lane  = t & 31;
    int ks    = (t >> 5) & 63;
    int chunk = t >> 11;
    int n  = chunk * 16 + (lane & 15);
    int d0 = ks * 4 + ((lane >> 4) << 1);
    v2f v;
    v.x = E[n * DIMS + d0];
    v.y = E[n * DIMS + d0 + 1];
    bp[t] = v;
}

// Fused fp32 GEMM (scores = x.E^T - 0.5||e||^2) + running argmax -> argmin(d2).
// One wave per 16-row tile; A tile resident in 128 VGPRs; B streamed from L2.
__global__ __launch_bounds__(256) void argmin_kernel(
    const float* __restrict__ x, const v2f* __restrict__ bp,
    const float* __restrict__ nh, int* __restrict__ idx_out) {
    int lane = threadIdx.x & 31;
    int tile = blockIdx.x * 8 + (threadIdx.x >> 5);     // 2048 tiles of 16 rows
    int row  = tile * 16 + (lane & 15);                 // flattened n = ((b*32+h)*32+w)
    int bimg = row >> 10;
    int hw   = row & 1023;
    const float* xb = x + (size_t)bimg * (DIMS * 1024) + hw;  // xf[n][d] = x[b][d][hw]
    int klane = (lane >> 4) << 1;                       // A 16x4 layout: lanes>=16 hold K={2,3}

    v2f a[64];
#pragma unroll
    for (int ks = 0; ks < 64; ++ks) {
        int d0 = ks * 4 + klane;
        v2f t;
        t.x = xb[(size_t)d0 * 1024];
        t.y = xb[(size_t)(d0 + 1) * 1024];
        a[ks] = t;
    }

    int col = lane & 15;
    float bestv[8];
    int   besti[8];
#pragma unroll
    for (int r = 0; r < 8; ++r) { bestv[r] = -3.4e38f; besti[r] = 0; }

    for (int chunk = 0; chunk < 64; ++chunk) {
        float nhv = nh[chunk * 16 + col];
        v8f c = {nhv, nhv, nhv, nhv, nhv, nhv, nhv, nhv};  // bias C with -0.5||e||^2
        const v2f* b = bp + chunk * (64 * 32) + lane;
#pragma unroll
        for (int ks = 0; ks < 64; ++ks) {
            v2f bb = b[ks * 32];
            c = __builtin_amdgcn_wmma_f32_16x16x4_f32(
                    false, a[ks], false, bb, (short)0, c, false, false);
        }
        int kidx = chunk * 16 + col;
#pragma unroll
        for (int r = 0; r < 8; ++r) {                    // strict > keeps lowest k on ties
            float v = c[r];
            if (v > bestv[r]) { bestv[r] = v; besti[r] = kidx; }
        }
    }
    // reduce argmax across the 16-lane column group (C layout: VGPR r = row r / r+8)
#pragma unroll
    for (int off = 1; off <= 8; off <<= 1) {
#pragma unroll
        for (int r = 0; r < 8; ++r) {
            float ov = __shfl_xor(bestv[r], off, 32);
            int   oi = __shfl_xor(besti[r], off, 32);
            if (ov > bestv[r] || (ov == bestv[r] && oi < besti[r])) { bestv[r] = ov; besti[r] = oi; }
        }
    }
    if ((lane & 15) == 0) {
        int mbase = tile * 16 + ((lane >> 4) << 3);
#pragma unroll
        for (int r = 0; r < 8; ++r) idx_out[mbase + r] = besti[r];
    }
}

__global__ void zero_enc_kernel(float* __restrict__ enc) {
    size_t i = ((size_t)blockIdx.x * blockDim.x + threadIdx.x) * 4;
    v4f z = {0.f, 0.f, 0.f, 0.f};
    *(v4f*)(enc + i) = z;
}

__global__ void zero_acc_kernel(unsigned long long* __restrict__ acc,
                                unsigned int* __restrict__ counts) {
    int i = blockIdx.x * blockDim.x + threadIdx.x;      // 262144
    acc[i] = 0ull;
    if (i < K_CLUSTERS) counts[i] = 0u;
}

__global__ void scatter_enc_kernel(const int* __restrict__ idx, float* __restrict__ enc,
                                   unsigned int* __restrict__ counts) {
    int n = blockIdx.x * blockDim.x + threadIdx.x;      // 32768
    int k = idx[n];
    enc[(size_t)n * K_CLUSTERS + k] = 1.0f;
    atomicAdd(&counts[k], 1u);
}

// quantized[b][d][hw] = E[idx[n]][d]; also fixed-point scatter of x into EMA accumulators.
__global__ void quant_scatter_kernel(const float* __restrict__ x, const float* __restrict__ E,
                                     const int* __restrict__ idx, float* __restrict__ q,
                                     unsigned long long* __restrict__ acc) {
    int b = blockIdx.x >> 8;
    int d = blockIdx.x & 255;
    size_t base = (size_t)b * (DIMS * 1024) + (size_t)d * 1024;
#pragma unroll
    for (int i = 0; i < 4; ++i) {
        int hw = threadIdx.x + i * 256;
        int n  = b * 1024 + hw;
        int k  = idx[n];
        float xv = x[base + hw];
        q[base + hw] = E[k * DIMS + d];
        long long f = (long long)llrintf(xv * FP_SCALE);
        atomicAdd(&acc[(size_t)k * DIMS + d], (unsigned long long)f);
    }
}

__global__ void newcs_kernel(const float* __restrict__ cs_in,
                             const unsigned int* __restrict__ counts,
                             float* __restrict__ cs_out) {
    __shared__ float sm[K_CLUSTERS];
    int k = threadIdx.x;
    float raw = DECAY * cs_in[k] + ONE_MINUS_DECAY * (float)counts[k];
    sm[k] = raw;
    __syncthreads();
    for (int s = 512; s > 0; s >>= 1) {
        if (k < s) sm[k] += sm[k + s];
        __syncthreads();
    }
    float tot = sm[0];
    cs_out[k] = (raw + EPSV) / (tot + K_CLUSTERS * EPSV) * tot;
}

__global__ void finalize_kernel(const float* __restrict__ ema_in,
                                const unsigned long long* __restrict__ acc,
                                const float* __restrict__ cs_out,
                                float* __restrict__ ema_out, float* __restrict__ emb_out) {
    int i = blockIdx.x * blockDim.x + threadIdx.x;      // 262144
    int k = i >> 8;
    float s = (float)(long long)acc[i] * INV_FP_SCALE;
    float e = DECAY * ema_in[i] + ONE_MINUS_DECAY * s;
    ema_out[i] = e;
    emb_out[i] = e / cs_out[k];
}

extern "C" void kernel_launch(void* const* d_in, const int* in_sizes, int n_in,
                              void* d_out, int out_size, void* d_ws, size_t ws_size,
                              hipStream_t stream) {
    const float* x   = (const float*)d_in[0];
    const float* E   = (const float*)d_in[1];
    const float* cs  = (const float*)d_in[2];
    const float* ema = (const float*)d_in[3];
    float* out = (float*)d_out;
    char*  ws  = (char*)d_ws;

    float*              nh     = (float*)(ws + WS_NH);
    v2f*                bp     = (v2f*)(ws + WS_BPERM);
    int*                idx    = (int*)(ws + WS_IDX);
    unsigned int*       counts = (unsigned int*)(ws + WS_COUNTS);
    unsigned long long* acc    = (unsigned long long*)(ws + WS_ACC);

    nh_e2_kernel<<<128, 256, 0, stream>>>(E, nh);
    bperm_kernel<<<512, 256, 0, stream>>>(E, bp);
    zero_acc_kernel<<<1024, 256, 0, stream>>>(acc, counts);
    zero_enc_kernel<<<32768, 256, 0, stream>>>(out + OUT_ENC);
    argmin_kernel<<<256, 256, 0, stream>>>(x, bp, nh, idx);
    scatter_enc_kernel<<<128, 256, 0, stream>>>(idx, out + OUT_ENC, counts);
    quant_scatter_kernel<<<8192, 256, 0, stream>>>(x, E, idx, out + OUT_QUANT, acc);
    newcs_kernel<<<1, 1024, 0, stream>>>(cs, counts, out + OUT_CS);
    finalize_kernel<<<1024, 256, 0, stream>>>(ema, acc, out + OUT_CS, out + OUT_EMA, out + OUT_EMB);
}